// TransformerBlockQuantum_65481071395463
// MI455X (gfx1250) — compile-verified
//
#include <hip/hip_runtime.h>
#include <math.h>

// Problem constants (match reference)
#define B_    4
#define S_    4096
#define E_    64
#define DK_   4
#define FFN_  256
#define NQ_   8
#define LN_EPS 1e-5f

typedef float v2f __attribute__((ext_vector_type(2)));
typedef float v8f __attribute__((ext_vector_type(8)));

// D(16x16,f32) = A(16x4,f32) x B(4x16,f32) + C  -- native CDNA5 fp32 WMMA
__device__ __forceinline__ v8f wmma4(v2f a, v2f b, v8f c) {
  return __builtin_amdgcn_wmma_f32_16x16x4_f32(false, a, false, b, (short)0, c, false, false);
}

__device__ __forceinline__ void lds_fence() {
  asm volatile("s_wait_dscnt 0x0" ::: "memory");
}

// ---------------------------------------------------------------------------
// Kernel 1: q4[b,s,:] = cos(x[b,s,0:4] + theta_attn)
// ---------------------------------------------------------------------------
__global__ void qfeat_kernel(const float* __restrict__ x,
                             const float* __restrict__ theta,
                             float* __restrict__ q4) {
  int idx = blockIdx.x * blockDim.x + threadIdx.x;
  if (idx >= B_ * S_) return;
  float4 xin = *(const float4*)(x + (size_t)idx * E_);
  float4 th  = *(const float4*)theta;
  float4 q;
  q.x = __cosf(xin.x + th.x);
  q.y = __cosf(xin.y + th.y);
  q.z = __cosf(xin.z + th.z);
  q.w = __cosf(xin.w + th.w);
  *(float4*)(q4 + (size_t)idx * 4) = q;
}

// ---------------------------------------------------------------------------
// Kernel 2: attention, head-dim 4, scale 8 folded into A.
// Scores are bounded in [-32,32] (cos features), so exp() is fp32-safe with
// NO max subtraction: unnormalized softmax, normalize once at the end.
//   - scores tile:  1x v_wmma_f32_16x16x4_f32
//   - p = exp(d) applied directly in D-layout registers
//   - psum += p  (element-wise; reduced across 16 lanes at the end)
//   - P@K: transpose-bounce P through LDS, 4x chained WMMA into persistent C
// grid = (S/64, B), block = 128 (4 waves). Each wave: 16 query rows.
// LDS: keys 64KB (whole batch) + 1KB P-scratch per wave.
// ---------------------------------------------------------------------------
__global__ void attn_kernel(const float* __restrict__ q4,
                            float* __restrict__ attn4) {
  extern __shared__ float smem[];
  float* keys = smem; // S_*4 floats
  const int b   = blockIdx.y;
  const int tid = threadIdx.x;

  for (int i = tid; i < S_; i += blockDim.x)
    *(float4*)&keys[i * 4] = *(const float4*)&q4[((size_t)b * S_ + i) * 4];
  __syncthreads();

  const int wave = tid >> 5;
  const int lane = tid & 31;
  const int lp   = lane & 15;
  const int half = lane >> 4;
  float* scratch = smem + S_ * 4 + wave * 256;
  const int rowBase = blockIdx.x * 64 + wave * 16;

  // A tile: lane holds (M=lp, K=2*half..2*half+1), pre-scaled by 8 (=16/sqrt(4))
  v2f a = *(const v2f*)&keys[(rowBase + lp) * 4 + 2 * half];
  a *= 8.0f;

  v8f accC = {}; // P @ K accumulator (columns 0..3 valid)
  v8f psum = {}; // unnormalized row-sum partials, per (row, col-lane)

  for (int kt = 0; kt < S_ / 16; ++kt) {
    // scores tile (already x8)
    v2f bv = *(const v2f*)&keys[(kt * 16 + lp) * 4 + 2 * half];
    v8f z = {};
    v8f d = wmma4(a, bv, z);

    // p = exp(scores), directly on the D-tile registers
    v8f p;
#pragma unroll
    for (int r = 0; r < 8; ++r) p[r] = __expf(d[r]);
    psum += p;

    // transpose-bounce P into A-layout via LDS (row = r+8*half, col = lp)
#pragma unroll
    for (int r = 0; r < 8; ++r)
      scratch[(r + 8 * half) * 16 + lp] = p[r];
    lds_fence();

    // acc += P_tile(16x16) @ K_tile(16x4), as 4 chained 16x16x4 WMMAs
#pragma unroll
    for (int j = 0; j < 4; ++j) {
      v2f av = *(const v2f*)&scratch[lp * 16 + 4 * j + 2 * half];
      const float* kb = &keys[(kt * 16 + 4 * j + 2 * half) * 4];
      v2f bk;
      bk.x = kb[lp];        // B(k=2*half  , n=lp)  (n>=4 columns are don't-care)
      bk.y = kb[16 + lp];   // B(k=2*half+1, n=lp)
      accC = wmma4(av, bk, accC);
    }
  }

  // row sums: reduce psum over the 16 lanes of each half (masks stay in-group)
  float inv[8];
#pragma unroll
  for (int r = 0; r < 8; ++r) {
    float s = psum[r];
    s += __shfl_xor(s, 1, 32);
    s += __shfl_xor(s, 2, 32);
    s += __shfl_xor(s, 4, 32);
    s += __shfl_xor(s, 8, 32);
    inv[r] = 1.0f / s;
  }

  // accC[r] at lane: row = r+8*half, k-component = lp (valid for lp<4)
  if (lp < 4) {
#pragma unroll
    for (int r = 0; r < 8; ++r)
      attn4[((size_t)b * S_ + rowBase + r + 8 * half) * 4 + lp] =
          accC[r] * inv[r];
  }
}

// ---------------------------------------------------------------------------
// Kernel 3: attn2[b,i,j] = attn4[b, (i*64+j)%4096, ((i*64+j)/4096)%4]
// (tile-16 broadcast + torch transpose(1,2).view fused into one gather)
// ---------------------------------------------------------------------------
__global__ void expand_kernel(const float* __restrict__ attn4,
                              float* __restrict__ attn2) {
  int idx = blockIdx.x * blockDim.x + threadIdx.x; // < B*S*E = 1M
  int b   = idx >> 18;            // / (S_*E_)
  int rem = idx & (S_ * E_ - 1);  // flat = i*64+j
  int sp  = rem & (S_ - 1);
  int c   = (rem >> 12) & 3;
  attn2[idx] = attn4[(((size_t)b << 12) + sp) * 4 + c];
}

// ---------------------------------------------------------------------------
// Kernel 4: x1 = LN1(x + attn2 @ Wc^T).  One wave per 16-row tile.
// WMMA fp32, K=64 in 16 chunks of 4; D-tiles bounced via LDS for row-wise LN.
// ---------------------------------------------------------------------------
__global__ void combine_ln1_kernel(const float* __restrict__ attn2,
                                   const float* __restrict__ Wc,
                                   const float* __restrict__ x,
                                   const float* __restrict__ g1,
                                   const float* __restrict__ b1,
                                   float* __restrict__ x1) {
  extern __shared__ float smem[];
  const int tid  = threadIdx.x;
  const int wave = tid >> 5, lane = tid & 31, lp = lane & 15, half = lane >> 4;
  float* rowbuf  = smem + wave * (16 * 64);
  const int g0   = (blockIdx.x * 4 + wave) * 16;

  for (int nt = 0; nt < 4; ++nt) {
    v8f c = {};
#pragma unroll
    for (int kc = 0; kc < 16; ++kc) {
      v2f av = *(const v2f*)&attn2[(size_t)(g0 + lp) * 64 + 4 * kc + 2 * half];
      v2f bv = *(const v2f*)&Wc[(nt * 16 + lp) * 64 + 4 * kc + 2 * half];
      c = wmma4(av, bv, c);
    }
#pragma unroll
    for (int r = 0; r < 8; ++r)
      rowbuf[(r + 8 * half) * 64 + nt * 16 + lp] = c[r];
  }
  lds_fence();

  const int g = g0 + lp;
  const float* xr = x + (size_t)g * 64 + half * 32;
  float4 v[8];
  float sum = 0.f, ss = 0.f;
#pragma unroll
  for (int q = 0; q < 8; ++q) {
    float4 t  = *(float4*)&rowbuf[lp * 64 + half * 32 + q * 4];
    float4 xv = *(const float4*)&xr[q * 4];
    t.x += xv.x; t.y += xv.y; t.z += xv.z; t.w += xv.w;
    v[q] = t;
    sum += t.x + t.y + t.z + t.w;
    ss  += t.x * t.x + t.y * t.y + t.z * t.z + t.w * t.w;
  }
  sum += __shfl_xor(sum, 16, 32);
  ss  += __shfl_xor(ss, 16, 32);
  float mean = sum * (1.f / 64.f);
  float var  = ss * (1.f / 64.f) - mean * mean;
  float rstd = rsqrtf(var + LN_EPS);
  float* orow = x1 + (size_t)g * 64 + half * 32;
#pragma unroll
  for (int q = 0; q < 8; ++q) {
    int col = half * 32 + q * 4;
    float4 gv = *(const float4*)&g1[col];
    float4 bv = *(const float4*)&b1[col];
    float4 o;
    o.x = (v[q].x - mean) * rstd * gv.x + bv.x;
    o.y = (v[q].y - mean) * rstd * gv.y + bv.y;
    o.z = (v[q].z - mean) * rstd * gv.z + bv.z;
    o.w = (v[q].w - mean) * rstd * gv.w + bv.w;
    *(float4*)&orow[q * 4] = o;
  }
}

// ---------------------------------------------------------------------------
// Kernel 5: out = LN2(x1 + relu(cos(x1[:, :8])*cos(th) @ W1^T) @ W2^T)
// One wave per 16-row tile. h (16x256) staged in LDS; GEMM K=256 via WMMA.
// ---------------------------------------------------------------------------
__global__ void ffn_ln2_kernel(const float* __restrict__ x1,
                               const float* __restrict__ theta,
                               const float* __restrict__ W1,
                               const float* __restrict__ W2,
                               const float* __restrict__ g2,
                               const float* __restrict__ b2,
                               float* __restrict__ out) {
  extern __shared__ float smem[];
  const int tid  = threadIdx.x;
  const int wave = tid >> 5, lane = tid & 31, lp = lane & 15, half = lane >> 4;
  float* qout   = smem + wave * 5248;  // 16*8
  float* h      = qout + 128;          // 16*256
  float* rowbuf = h + 4096;            // 16*64
  const int g0  = (blockIdx.x * 4 + wave) * 16;

  { // q_out = cos(x1[:, :8]) * cos(theta_ffn)   (lane -> row lp, cols half*4..)
    float4 xv = *(const float4*)&x1[(size_t)(g0 + lp) * 64 + half * 4];
    float4 tv = *(const float4*)&theta[half * 4];
    float4 q;
    q.x = __cosf(xv.x) * __cosf(tv.x);
    q.y = __cosf(xv.y) * __cosf(tv.y);
    q.z = __cosf(xv.z) * __cosf(tv.z);
    q.w = __cosf(xv.w) * __cosf(tv.w);
    *(float4*)&qout[lp * 8 + half * 4] = q;
  }
  lds_fence();

  // h = relu(q_out @ W1^T), 16x256
  for (int idx = lane; idx < 16 * 256; idx += 32) {
    int r = idx >> 8, f = idx & 255;
    float4 qa = *(float4*)&qout[r * 8];
    float4 qb = *(float4*)&qout[r * 8 + 4];
    float4 wa = *(const float4*)&W1[f * 8];
    float4 wb = *(const float4*)&W1[f * 8 + 4];
    float s = qa.x * wa.x + qa.y * wa.y + qa.z * wa.z + qa.w * wa.w +
              qb.x * wb.x + qb.y * wb.y + qb.z * wb.z + qb.w * wb.w;
    h[r * 256 + f] = fmaxf(s, 0.f);
  }
  lds_fence();

  // ffn = h @ W2^T via WMMA, K=256 in 64 chunks
  for (int nt = 0; nt < 4; ++nt) {
    v8f c = {};
    for (int kc = 0; kc < 64; ++kc) {
      v2f av = *(const v2f*)&h[lp * 256 + 4 * kc + 2 * half];
      v2f bv = *(const v2f*)&W2[(nt * 16 + lp) * 256 + 4 * kc + 2 * half];
      c = wmma4(av, bv, c);
    }
#pragma unroll
    for (int r = 0; r < 8; ++r)
      rowbuf[(r + 8 * half) * 64 + nt * 16 + lp] = c[r];
  }
  lds_fence();

  // LN2 with residual x1
  const int g = g0 + lp;
  const float* xr = x1 + (size_t)g * 64 + half * 32;
  float4 v[8];
  float sum = 0.f, ss = 0.f;
#pragma unroll
  for (int q = 0; q < 8; ++q) {
    float4 t  = *(float4*)&rowbuf[lp * 64 + half * 32 + q * 4];
    float4 xv = *(const float4*)&xr[q * 4];
    t.x += xv.x; t.y += xv.y; t.z += xv.z; t.w += xv.w;
    v[q] = t;
    sum += t.x + t.y + t.z + t.w;
    ss  += t.x * t.x + t.y * t.y + t.z * t.z + t.w * t.w;
  }
  sum += __shfl_xor(sum, 16, 32);
  ss  += __shfl_xor(ss, 16, 32);
  float mean = sum * (1.f / 64.f);
  float var  = ss * (1.f / 64.f) - mean * mean;
  float rstd = rsqrtf(var + LN_EPS);
  float* orow = out + (size_t)g * 64 + half * 32;
#pragma unroll
  for (int q = 0; q < 8; ++q) {
    int col = half * 32 + q * 4;
    float4 gv = *(const float4*)&g2[col];
    float4 bv = *(const float4*)&b2[col];
    float4 o;
    o.x = (v[q].x - mean) * rstd * gv.x + bv.x;
    o.y = (v[q].y - mean) * rstd * gv.y + bv.y;
    o.z = (v[q].z - mean) * rstd * gv.z + bv.z;
    o.w = (v[q].w - mean) * rstd * gv.w + bv.w;
    *(float4*)&orow[q * 4] = o;
  }
}

// ---------------------------------------------------------------------------
extern "C" void kernel_launch(void* const* d_in, const int* in_sizes, int n_in,
                              void* d_out, int out_size, void* d_ws, size_t ws_size,
                              hipStream_t stream) {
  const float* x          = (const float*)d_in[0];
  const float* theta_attn = (const float*)d_in[1];
  const float* theta_ffn  = (const float*)d_in[2];
  const float* W_combine  = (const float*)d_in[3];
  const float* W1         = (const float*)d_in[4];
  const float* W2         = (const float*)d_in[5];
  const float* ln1_g      = (const float*)d_in[6];
  const float* ln1_b      = (const float*)d_in[7];
  const float* ln2_g      = (const float*)d_in[8];
  const float* ln2_b      = (const float*)d_in[9];
  float* out = (float*)d_out;

  float* ws    = (float*)d_ws;
  float* q4    = ws;                           // B*S*4   = 65536
  float* attn4 = q4 + (size_t)B_ * S_ * 4;     // B*S*4   = 65536
  float* attn2 = attn4 + (size_t)B_ * S_ * 4;  // B*S*64  = 1048576
  float* x1    = attn2 + (size_t)B_ * S_ * E_; // B*S*64  = 1048576

  // 1) quantum features for attention
  qfeat_kernel<<<(B_ * S_ + 255) / 256, 256, 0, stream>>>(x, theta_attn, q4);

  // 2) attention (WMMA f32 16x16x4 for both QK^T and P@K), keys resident in LDS
  {
    dim3 grid(S_ / 64, B_);
    size_t lds = (size_t)(S_ * 4 + 4 * 256) * sizeof(float); // 68 KB
    attn_kernel<<<grid, 128, lds, stream>>>(q4, attn4);
  }

  // 3) tile-broadcast + transpose gather
  expand_kernel<<<(B_ * S_ * E_) / 256, 256, 0, stream>>>(attn4, attn2);

  // 4) combine matmul + residual + LN1
  {
    size_t lds = (size_t)4 * 16 * 64 * sizeof(float); // 16 KB
    combine_ln1_kernel<<<(B_ * S_ / 16) / 4, 128, lds, stream>>>(
        attn2, W_combine, x, ln1_g, ln1_b, x1);
  }

  // 5) quantum FFN + residual + LN2
  {
    size_t lds = (size_t)4 * 5248 * sizeof(float); // 82 KB
    ffn_ln2_kernel<<<(B_ * S_ / 16) / 4, 128, lds, stream>>>(
        x1, theta_ffn, W1, W2, ln2_g, ln2_b, out);
  }
}